// HGTLayer_29145648070964
// MI455X (gfx1250) — compile-verified
//
#include <hip/hip_runtime.h>
#include <math.h>
#include <stdint.h>

typedef float v2f __attribute__((ext_vector_type(2)));
typedef float v8f __attribute__((ext_vector_type(8)));

#define H   4
#define DK  32
#define DIM 128

// ---------- order-preserving float<->uint key for atomicMax-based segment max ----------
__device__ __forceinline__ unsigned f2key(float f) {
    unsigned u = __float_as_uint(f);
    return (u & 0x80000000u) ? ~u : (u | 0x80000000u);
}
__device__ __forceinline__ float key2f(unsigned k) {
    unsigned u = (k & 0x80000000u) ? (k & 0x7FFFFFFFu) : ~k;
    return __uint_as_float(u);
}

// ---------- fold per-head 32x32 relation transforms into 128x128 projection weights ----
// Wc[e][h*32+j][d] = sum_i R[e][h][i][j] * W[t(e)][h*32+i][d]   (and same for bias)
__global__ void build_combined(const float* __restrict__ Wk, const float* __restrict__ bk,
                               const float* __restrict__ Wv, const float* __restrict__ bv,
                               const float* __restrict__ rel_att, const float* __restrict__ rel_msg,
                               float* __restrict__ WcK, float* __restrict__ WcV,
                               float* __restrict__ bcK, float* __restrict__ bcV)
{
    int idx = blockIdx.x * blockDim.x + threadIdx.x;   // 2 * 3 * 128 * 128 threads
    if (idx >= 2 * 3 * DIM * DIM) return;
    int kv = (idx >= 3 * DIM * DIM) ? 1 : 0;
    int r  = idx - kv * 3 * DIM * DIM;
    int e      = r >> 14;          // relation 0..2
    int jprime = (r >> 7) & 127;   // output row
    int d      = r & 127;          // input col
    int h = jprime >> 5, j = jprime & 31;
    int t = (e == 1) ? 1 : 0;      // src node type: cites=paper, writes=author, rev=paper
    const float* W = kv ? Wv : Wk;
    const float* B = kv ? bv : bk;
    const float* R = kv ? rel_msg : rel_att;
    float acc = 0.f;
    #pragma unroll 8
    for (int i = 0; i < 32; ++i)
        acc += W[(t * DIM + h * 32 + i) * DIM + d] * R[((e * H + h) * 32 + i) * 32 + j];
    (kv ? WcV : WcK)[e * DIM * DIM + jprime * DIM + d] = acc;
    if (d == 0) {
        float bb = 0.f;
        for (int i = 0; i < 32; ++i)
            bb += B[t * DIM + h * 32 + i] * R[((e * H + h) * 32 + i) * 32 + j];
        (kv ? bcV : bcK)[e * DIM + jprime] = bb;
    }
}

// ---------- WMMA fp32 GEMM: C[N,128] = ascale*(A[N,128] @ W^T) + bias, optional skip-mix
// 256-thread block = 8 waves x 16 rows. W staged once per block into LDS via CDNA5
// async global->LDS copies (ASYNCcnt); B fragments then come from conflict-free LDS.
__global__ __launch_bounds__(256) void gemm128(const float* __restrict__ A,
                                               const float* __restrict__ W,
                                               const float* __restrict__ bias,
                                               float* __restrict__ C, int N,
                                               float ascale,
                                               const float* __restrict__ skipv,
                                               const float* __restrict__ hres)
{
    constexpr int WPAD = DIM + 4;   // 132-float row stride: 16B aligned, bank-conflict-free
    __shared__ __align__(16) float Ws[DIM * WPAD];   // 66 KB of the WGP's 320 KB LDS

    // --- stage W[128][128] into LDS: each lane moves one 16B chunk, no VGPR staging ---
    unsigned lds_base = (unsigned)(uintptr_t)(&Ws[0]);
    for (int i = threadIdx.x; i < DIM * (DIM / 4); i += 256) {
        int row = i >> 5;                 // 32 float4 per source row
        int c4  = (i & 31) << 2;
        unsigned loff = lds_base + (unsigned)(row * WPAD + c4) * 4u;
        unsigned long long gp = (unsigned long long)(uintptr_t)(W + row * DIM + c4);
        asm volatile("global_load_async_to_lds_b128 %0, %1, off"
                     :: "v"(loff), "v"(gp) : "memory");
    }
    asm volatile("s_wait_asynccnt 0x0" ::: "memory");
    __syncthreads();

    const int lane = threadIdx.x & 31;       // wave32
    const int wv   = threadIdx.x >> 5;       // wave id: 8 waves per block
    const int m    = lane & 15;              // row/col within 16x16 tile
    const int ks   = (lane >> 4) << 1;       // lanes 0-15: K=0,1 ; lanes 16-31: K=2,3
    const int row0 = blockIdx.x * 128 + wv * 16;
    int ar = row0 + m; if (ar >= N) ar = N - 1;      // clamp (keeps EXEC all-ones)
    const float* Arow = A + (size_t)ar * DIM + ks;

    v8f acc[8];
    #pragma unroll
    for (int t = 0; t < 8; ++t) acc[t] = (v8f){0.f,0.f,0.f,0.f,0.f,0.f,0.f,0.f};

    for (int kk = 0; kk < DIM; kk += 4) {
        v2f a = *(const v2f*)(Arow + kk);
        #pragma unroll
        for (int t = 0; t < 8; ++t) {
            // B[k][n] = W[n][k] : per-lane float2 from LDS row (tile col), offset kk+ks
            v2f b = *(const v2f*)(Ws + (size_t)(t * 16 + m) * WPAD + kk + ks);
            acc[t] = __builtin_amdgcn_wmma_f32_16x16x4_f32(
                false, a, false, b, (short)0, acc[t], false, false);
        }
    }

    float alpha = 1.f, beta = 0.f;
    if (skipv) { float s = *skipv; alpha = 1.f / (1.f + __expf(-s)); beta = 1.f - alpha; }
    const int mbase = (lane >> 4) << 3;      // C/D: reg r -> M = r + 8*(lane>=16)
    #pragma unroll
    for (int t = 0; t < 8; ++t) {
        int col = t * 16 + m;
        float bval = bias ? bias[col] : 0.f;
        #pragma unroll
        for (int r = 0; r < 8; ++r) {
            int row = row0 + r + mbase;
            if (row < N) {
                float val = acc[t][r] * ascale + bval;
                if (skipv) val = alpha * val + beta * hres[(size_t)row * DIM + col];
                C[(size_t)row * DIM + col] = val;
            }
        }
    }
}

// ---------- utility fill ----------
__global__ void fill_u32(unsigned* __restrict__ p, int n, unsigned v) {
    int i = blockIdx.x * blockDim.x + threadIdx.x;
    if (i < n) p[i] = v;
}

// ---------- pass 1: per-(edge,head) score -> store + atomic segment max ----------
__global__ void score_max(const int* __restrict__ src, const int* __restrict__ dst, int E,
                          const float* __restrict__ q, const float* __restrict__ k,
                          const float* __restrict__ pri, float inv_sqrt_dk,
                          float* __restrict__ score, unsigned* __restrict__ mbuf)
{
    int tid = blockIdx.x * blockDim.x + threadIdx.x;
    if (tid >= E * H) return;
    int e = tid >> 2, h = tid & 3;
    int s = src[e], d = dst[e];
    const float* qp = q + (size_t)d * DIM + h * DK;
    const float* kp = k + (size_t)s * DIM + h * DK;
    float acc = 0.f;
    #pragma unroll
    for (int i = 0; i < DK; i += 4) {
        float4 qv = *(const float4*)(qp + i);
        float4 kv = *(const float4*)(kp + i);
        acc += qv.x * kv.x + qv.y * kv.y + qv.z * kv.z + qv.w * kv.w;
    }
    float sc = acc * pri[h] * inv_sqrt_dk;
    score[tid] = sc;
    atomicMax(&mbuf[(size_t)d * H + h], f2key(sc));
}

// ---------- pass 2: e = exp(s - m), overwrite score, atomic sum into z ----------
__global__ void exp_sum(const int* __restrict__ dst, int E,
                        float* __restrict__ score, const unsigned* __restrict__ mbuf,
                        float* __restrict__ z)
{
    int tid = blockIdx.x * blockDim.x + threadIdx.x;
    if (tid >= E * H) return;
    int e = tid >> 2, h = tid & 3;
    int d = dst[e];
    float mv = key2f(mbuf[(size_t)d * H + h]);
    float ev = __expf(score[tid] - mv);
    score[tid] = ev;
    atomicAdd(&z[(size_t)d * H + h], ev);
}

// ---------- pass 3: scatter normalized a * v into t (32 lanes = 128 dims per edge) ----
__global__ void scatter_msg(const int* __restrict__ src, const int* __restrict__ dst, int E,
                            const float* __restrict__ score, const float* __restrict__ z,
                            const float* __restrict__ v, float* __restrict__ t)
{
    int tid = blockIdx.x * blockDim.x + threadIdx.x;
    if (tid >= E * 32) return;
    int e = tid >> 5, l = tid & 31, h = l >> 3;
    int s = src[e], d = dst[e];
    float a = score[(size_t)e * H + h] / z[(size_t)d * H + h];
    float4 vv = *(const float4*)(v + (size_t)s * DIM + l * 4);
    float* tp = t + (size_t)d * DIM + l * 4;
    atomicAdd(tp + 0, a * vv.x);
    atomicAdd(tp + 1, a * vv.y);
    atomicAdd(tp + 2, a * vv.z);
    atomicAdd(tp + 3, a * vv.w);
}

extern "C" void kernel_launch(void* const* d_in, const int* in_sizes, int n_in,
                              void* d_out, int out_size, void* d_ws, size_t ws_size,
                              hipStream_t stream)
{
    (void)n_in; (void)out_size; (void)ws_size;
    const float* h_paper  = (const float*)d_in[0];
    const float* h_author = (const float*)d_in[1];
    const float* Wk = (const float*)d_in[2];
    const float* bk = (const float*)d_in[3];
    const float* Wq = (const float*)d_in[4];
    const float* bq = (const float*)d_in[5];
    const float* Wv = (const float*)d_in[6];
    const float* bv = (const float*)d_in[7];
    const float* Wa = (const float*)d_in[8];
    const float* ba = (const float*)d_in[9];
    const float* rel_att = (const float*)d_in[10];
    const float* rel_msg = (const float*)d_in[11];
    const float* rel_pri = (const float*)d_in[12];
    const float* skip    = (const float*)d_in[13];
    const int* cites_src  = (const int*)d_in[14];
    const int* cites_dst  = (const int*)d_in[15];
    const int* writes_src = (const int*)d_in[16];
    const int* writes_dst = (const int*)d_in[17];
    const int* rev_src    = (const int*)d_in[18];
    const int* rev_dst    = (const int*)d_in[19];

    const int NP = in_sizes[0] / DIM;
    const int NA = in_sizes[1] / DIM;
    const int EC = in_sizes[14];
    const int EW = in_sizes[16];
    const int ER = in_sizes[18];
    const int NMAX = NP > NA ? NP : NA;
    int EMAX = EC > EW ? EC : EW; if (ER > EMAX) EMAX = ER;
    const float inv_sqrt_dk = 0.17677669529663689f; // 1/sqrt(32)

    // ---- scratch layout (floats, 256B-aligned regions) ----
    float* base = (float*)d_ws;
    size_t off = 0;
    auto alloc = [&](size_t n) { float* p = base + off; off += (n + 63) & ~(size_t)63; return p; };
    float*    WcK   = alloc(3 * DIM * DIM);
    float*    WcV   = alloc(3 * DIM * DIM);
    float*    bcK   = alloc(3 * DIM);
    float*    bcV   = alloc(3 * DIM);
    float*    q_p   = alloc((size_t)NP * DIM);
    float*    q_a   = alloc((size_t)NA * DIM);
    float*    kbuf  = alloc((size_t)NMAX * DIM);
    float*    vbuf  = alloc((size_t)NMAX * DIM);
    float*    t_p   = alloc((size_t)NP * DIM);
    float*    t_a   = alloc((size_t)NA * DIM);
    unsigned* mbuf  = (unsigned*)alloc((size_t)NMAX * H);
    float*    zbuf  = alloc((size_t)NMAX * H);
    float*    score = alloc((size_t)EMAX * H);

    auto gemm = [&](const float* A, const float* W, const float* b, float* C, int N,
                    float ascale, const float* sk, const float* hres) {
        gemm128<<<(N + 127) / 128, 256, 0, stream>>>(A, W, b, C, N, ascale, sk, hres);
    };
    auto fillu = [&](void* p, size_t n, unsigned v) {
        fill_u32<<<((int)n + 255) / 256, 256, 0, stream>>>((unsigned*)p, (int)n, v);
    };

    // 1) fold relation transforms into projection weights
    build_combined<<<(2 * 3 * DIM * DIM + 255) / 256, 256, 0, stream>>>(
        Wk, bk, Wv, bv, rel_att, rel_msg, WcK, WcV, bcK, bcV);

    // 2) Q projections (plain GEMM with bias)
    gemm(h_paper,  Wq,              bq,       q_p, NP, 1.f, nullptr, nullptr);
    gemm(h_author, Wq + DIM * DIM,  bq + DIM, q_a, NA, 1.f, nullptr, nullptr);

    // 3) zero accumulators
    fillu(t_p, (size_t)NP * DIM, 0u);
    fillu(t_a, (size_t)NA * DIM, 0u);

    auto relation = [&](int rel, const float* hsrc, int Nsrc, const float* qdst, int Ndst,
                        const int* esrc, const int* edst, int E, float* tdst) {
        gemm(hsrc, WcK + (size_t)rel * DIM * DIM, bcK + rel * DIM, kbuf, Nsrc, 1.f, nullptr, nullptr);
        gemm(hsrc, WcV + (size_t)rel * DIM * DIM, bcV + rel * DIM, vbuf, Nsrc, 1.f, nullptr, nullptr);
        fillu(mbuf, (size_t)Ndst * H, 0u);   // key(-inf)-ish floor
        fillu(zbuf, (size_t)Ndst * H, 0u);   // 0.0f
        int T1 = E * H;
        score_max<<<(T1 + 255) / 256, 256, 0, stream>>>(esrc, edst, E, qdst, kbuf,
                                                        rel_pri + rel * H, inv_sqrt_dk, score, mbuf);
        exp_sum<<<(T1 + 255) / 256, 256, 0, stream>>>(edst, E, score, mbuf, zbuf);
        int T3 = E * 32;
        scatter_msg<<<(T3 + 255) / 256, 256, 0, stream>>>(esrc, edst, E, score, zbuf, vbuf, tdst);
    };

    relation(0, h_paper,  NP, q_p, NP, cites_src,  cites_dst,  EC, t_p);
    relation(1, h_author, NA, q_p, NP, writes_src, writes_dst, EW, t_p);
    relation(2, h_paper,  NP, q_a, NA, rev_src,    rev_dst,    ER, t_a);

    // 4) output GEMM with mean-scale + sigmoid-skip epilogue
    float* out = (float*)d_out;
    gemm(t_p, Wa,             ba,       out,                     NP, 0.5f, skip,     h_paper);
    gemm(t_a, Wa + DIM * DIM, ba + DIM, out + (size_t)NP * DIM, NA, 1.0f, skip + 1, h_author);
}